// HungarianMatcher_63273458204989
// MI455X (gfx1250) — compile-verified
//
#include <hip/hip_runtime.h>
#include <hip/hip_bf16.h>
#include <math.h>

// ---------------- problem constants ----------------
#define B_    2
#define Q_    250
#define QPAD_ 256
#define C_    134
#define CPAD_ 136          // 34 * 4 for WMMA 16x16x4 K-steps
#define T_    32
#define P_    12544
#define HP_   256
#define WP_   256
#define HG_   512
#define WG_   512

// pairwise-L1 tiling: uniform async staging (exactly 3 b128 copies/thread/chunk)
#define NS_   28                   // P slices (grid parallelism)
#define PS_   (P_ / NS_)           // 448 points per slice
#define CH_   64                   // points staged in LDS per chunk
#define NCH_  (PS_ / CH_)          // 7 chunks
#define CH4_  (CH_ / 4)            // 16 float4 per row-chunk
#define LDSTR_ 68                  // padded row stride: 272B = 16B-aligned, 4-bank rotation/row

typedef float v2f __attribute__((ext_vector_type(2)));
typedef float v8f __attribute__((ext_vector_type(8)));

// ---------------- workspace layout (float offsets) ----------------
#define WS_PROBS   0
#define WS_CLASS   (WS_PROBS   + B_*QPAD_*CPAD_)   // 69632
#define WS_OUTPTS  (WS_CLASS   + B_*QPAD_*T_)      // +16384
#define WS_TGTPTS  (WS_OUTPTS  + (size_t)B_*Q_*P_) // +6272000
#define WS_PARTIAL (WS_TGTPTS  + (size_t)B_*T_*P_) // +802816
// partial: NS_*B_*QPAD_*T_ = 458752 floats; total ~30.5 MB

// ---------------- 1) softmax into padded probs ----------------
__global__ void softmax_kernel(const float* __restrict__ logits,
                               float* __restrict__ probs) {
  int idx = blockIdx.x * blockDim.x + threadIdx.x;   // over B*QPAD
  if (idx >= B_ * QPAD_) return;
  int q = idx % QPAD_, b = idx / QPAD_;
  float* row = probs + (size_t)idx * CPAD_;
  if (q >= Q_) {                                     // zero pad rows 250..255
    for (int c = 0; c < CPAD_; ++c) row[c] = 0.f;
    return;
  }
  const float* in = logits + ((size_t)b * Q_ + q) * C_;
  float m = -INFINITY;
  for (int c = 0; c < C_; ++c) m = fmaxf(m, in[c]);
  float s = 0.f;
  for (int c = 0; c < C_; ++c) s += __expf(in[c] - m);
  float inv = 1.f / s;
  for (int c = 0; c < C_; ++c) row[c] = __expf(in[c] - m) * inv;
  row[C_] = 0.f; row[C_ + 1] = 0.f;                  // zero pad classes 134,135
}

// ---------------- 2) bilinear point sampler (grid_sample equivalent) ----------------
__global__ void sample_kernel(const float* __restrict__ masks,
                              const float* __restrict__ coords,
                              float* __restrict__ out,
                              int N, int H, int W, long long total) {
  long long idx = (long long)blockIdx.x * blockDim.x + threadIdx.x;
  if (idx >= total) return;
  int p  = (int)(idx % P_);
  int bn = (int)(idx / P_);
  int b  = bn / N;

  float x = coords[((size_t)b * P_ + p) * 2 + 0];
  float y = coords[((size_t)b * P_ + p) * 2 + 1];
  float ix = x * (float)W - 0.5f;
  float iy = y * (float)H - 0.5f;
  float x0f = floorf(ix), y0f = floorf(iy);
  float wx = ix - x0f, wy = iy - y0f;
  int x0 = (int)x0f, y0 = (int)y0f;
  int x1 = x0 + 1,   y1 = y0 + 1;

  const float* base = masks + (size_t)bn * H * W;
  int xc0 = min(max(x0, 0), W - 1), xc1 = min(max(x1, 0), W - 1);
  int yc0 = min(max(y0, 0), H - 1), yc1 = min(max(y1, 0), H - 1);
  float g00 = (x0 >= 0 && x0 < W && y0 >= 0 && y0 < H) ? base[(size_t)yc0 * W + xc0] : 0.f;
  float g01 = (x1 >= 0 && x1 < W && y0 >= 0 && y0 < H) ? base[(size_t)yc0 * W + xc1] : 0.f;
  float g10 = (x0 >= 0 && x0 < W && y1 >= 0 && y1 < H) ? base[(size_t)yc1 * W + xc0] : 0.f;
  float g11 = (x1 >= 0 && x1 < W && y1 >= 0 && y1 < H) ? base[(size_t)yc1 * W + xc1] : 0.f;

  out[idx] = g00 * (1.f - wx) * (1.f - wy) + g01 * wx * (1.f - wy)
           + g10 * (1.f - wx) * wy         + g11 * wx * wy;
}

// ---------------- 3) class cost via one-hot WMMA matmul ----------------
// classcost[b, q, t] = probs[b, q, label[b, t]]  ==  (probs · onehot(labels))[q, t]
#if __has_builtin(__builtin_amdgcn_wmma_f32_16x16x4_f32)
#define HAVE_WMMA_F32X4 1
#endif

__global__ void __launch_bounds__(32)
class_cost_kernel(const float* __restrict__ probs,
                  const int* __restrict__ labels,
                  float* __restrict__ classcost) {
  int bi = blockIdx.x;            // [b][qtile(16)][thalf(2)]
  int th = bi & 1;
  int qt = (bi >> 1) & 15;
  int b  = bi >> 5;
  int qbase = qt * 16, tbase = th * 16;
  int lane = threadIdx.x;         // one full wave32
  int h = lane >> 4;              // K-half selector
  int l = lane & 15;              // A: row M; B/D: col N

  int lab = labels[b * T_ + tbase + l];
  const float* prow = probs + ((size_t)(b * QPAD_ + qbase + l)) * CPAD_;

#ifdef HAVE_WMMA_F32X4
  v8f c = {0.f, 0.f, 0.f, 0.f, 0.f, 0.f, 0.f, 0.f};
  for (int s = 0; s < CPAD_ / 4; ++s) {             // 34 K-steps of 4
    int k0 = 4 * s;
    // A 16x4 f32: lanes0-15 -> {K0,K1}, lanes16-31 -> {K2,K3}
    v2f A;  A.x = prow[k0 + 2 * h];
            A.y = prow[k0 + 2 * h + 1];
    // B 4x16 one-hot: B[k][n] = (label[n] == k0+k)
    v2f Bm; Bm.x = (lab == k0 + 2 * h)     ? 1.f : 0.f;
            Bm.y = (lab == k0 + 2 * h + 1) ? 1.f : 0.f;
    c = __builtin_amdgcn_wmma_f32_16x16x4_f32(false, A, false, Bm,
                                              (short)0, c, false, false);
  }
#pragma unroll
  for (int r = 0; r < 8; ++r)   // D layout: VGPR r, lane-half h -> row r+8h, col l
    classcost[((size_t)(b * QPAD_ + qbase + r + 8 * h)) * T_ + tbase + l] = c[r];
#else
  for (int r = 0; r < 8; ++r) { // scalar fallback (exact gather)
    int q = qbase + r + 8 * h;
    classcost[((size_t)(b * QPAD_ + q)) * T_ + tbase + l] =
        probs[((size_t)(b * QPAD_ + q)) * CPAD_ + lab];
  }
#endif
}

// ---------------- 4) pairwise L1: double-buffered async global->LDS pipeline ----------
__global__ void __launch_bounds__(256)
pair_l1_kernel(const float* __restrict__ out_pts,
               const float* __restrict__ tgt_pts,
               float* __restrict__ partial) {
  __shared__ __align__(16) float sOut[2][16][LDSTR_];
  __shared__ __align__(16) float sTgt[2][32][LDSTR_];

  int bi = blockIdx.x;            // [s(NS)][b(2)][qtile(16)]
  int qt = bi & 15;
  int b  = (bi >> 4) & 1;
  int s  = bi >> 5;
  int qbase = qt * 16;
  int ps = s * PS_;
  int tid = threadIdx.x;
  int q  = tid & 15;              // lanes 0..15 of each wave: q ; broadcast-friendly t
  int tp = tid >> 4;              // 0..15 -> targets tp and tp+16

  // per-thread precomputed staging targets: exactly 3 b128 copies per chunk
  //   k=0: out tile  (16 rows x 16 float4)   i = tid          (0..255)
  //   k=1,2: tgt tile (32 rows x 16 float4)  j = tid + k*256 - 256 (0..511)
  unsigned ldsOffK[2][3];
  size_t   grow[3];
  {
    int r0 = tid >> 4, c40 = tid & 15;                 // out row/col4
    int qg = qbase + r0; if (qg > Q_ - 1) qg = Q_ - 1; // clamp padded q-rows
    grow[0] = ((size_t)(b * Q_ + qg)) * P_ + c40 * 4;
    ldsOffK[0][0] = (unsigned)(size_t)&sOut[0][r0][c40 * 4];
    ldsOffK[1][0] = (unsigned)(size_t)&sOut[1][r0][c40 * 4];
#pragma unroll
    for (int k = 1; k < 3; ++k) {
      int j = tid + k * 256 - 256;
      int r = j >> 4, c4 = j & 15;                     // tgt row/col4
      grow[k] = ((size_t)(b * T_ + r)) * P_ + c4 * 4;
      ldsOffK[0][k] = (unsigned)(size_t)&sTgt[0][r][c4 * 4];
      ldsOffK[1][k] = (unsigned)(size_t)&sTgt[1][r][c4 * 4];
    }
  }

#define STAGE(CHUNK, BUF)                                                    \
  do {                                                                       \
    int pbase_ = ps + (CHUNK) * CH_;                                         \
    _Pragma("unroll")                                                        \
    for (int k_ = 0; k_ < 3; ++k_) {                                         \
      const float* gp_ = (k_ == 0 ? out_pts : tgt_pts) + grow[k_] + pbase_;  \
      asm volatile("global_load_async_to_lds_b128 %0, %1, off"               \
                   :: "v"(ldsOffK[(BUF)][k_]), "v"(gp_) : "memory");         \
    }                                                                        \
  } while (0)

  float4 a0 = make_float4(0.f, 0.f, 0.f, 0.f);
  float4 a1 = make_float4(0.f, 0.f, 0.f, 0.f);

  STAGE(0, 0);
  for (int ch = 0; ch < NCH_; ++ch) {
    int buf = ch & 1;
    if (ch + 1 < NCH_) {
      STAGE(ch + 1, buf ^ 1);
      // only the newest batch (3 copies) may remain outstanding -> chunk `ch` landed
      asm volatile("s_wait_asynccnt 0x3" ::: "memory");
    } else {
      asm volatile("s_wait_asynccnt 0x0" ::: "memory");
    }
    __syncthreads();              // all waves' chunk-`ch` LDS writes visible

    const float4* orow = (const float4*)&sOut[buf][q][0];
    const float4* t0r  = (const float4*)&sTgt[buf][tp][0];
    const float4* t1r  = (const float4*)&sTgt[buf][tp + 16][0];
#pragma unroll 4
    for (int c4 = 0; c4 < CH4_; ++c4) {
      float4 o = orow[c4], u = t0r[c4], v = t1r[c4];
      a0.x += fabsf(o.x - u.x); a0.y += fabsf(o.y - u.y);
      a0.z += fabsf(o.z - u.z); a0.w += fabsf(o.w - u.w);
      a1.x += fabsf(o.x - v.x); a1.y += fabsf(o.y - v.y);
      a1.z += fabsf(o.z - v.z); a1.w += fabsf(o.w - v.w);
    }
    __syncthreads();              // done reading `buf` before it is re-staged (ch+2)
  }
#undef STAGE

  float r0 = (a0.x + a0.y) + (a0.z + a0.w);
  float r1 = (a1.x + a1.y) + (a1.z + a1.w);
  size_t base = ((size_t)((s * B_ + b) * QPAD_ + qbase + q)) * T_;
  partial[base + tp]      = r0;   // each (s,b,q,t) written exactly once: deterministic
  partial[base + tp + 16] = r1;
}

// ---------------- 5) finalize: 10*mean_L1 + 2*(-prob), NaN -> 0 ----------------
__global__ void finalize_kernel(const float* __restrict__ partial,
                                const float* __restrict__ classcost,
                                float* __restrict__ out) {
  int idx = blockIdx.x * blockDim.x + threadIdx.x;
  if (idx >= B_ * Q_ * T_) return;
  int t = idx % T_;
  int q = (idx / T_) % Q_;
  int b = idx / (T_ * Q_);
  float m = 0.f;
  for (int s = 0; s < NS_; ++s)
    m += partial[((size_t)((s * B_ + b) * QPAD_ + q)) * T_ + t];
  float cc = classcost[((size_t)(b * QPAD_ + q)) * T_ + t];
  float r = 10.f * (m / (float)P_) - 2.f * cc;
  out[idx] = (r != r) ? 0.f : r;
}

// ---------------- launch ----------------
extern "C" void kernel_launch(void* const* d_in, const int* in_sizes, int n_in,
                              void* d_out, int out_size, void* d_ws, size_t ws_size,
                              hipStream_t stream) {
  (void)in_sizes; (void)n_in; (void)out_size; (void)ws_size;
  const float* pred_logits = (const float*)d_in[0];
  const float* pred_masks  = (const float*)d_in[1];
  const float* tgt_masks   = (const float*)d_in[2];
  const float* coords      = (const float*)d_in[3];
  const int*   labels      = (const int*)d_in[4];   // JAX default x64-off -> int32

  float* ws      = (float*)d_ws;
  float* probs   = ws + WS_PROBS;
  float* classc  = ws + WS_CLASS;
  float* out_pts = ws + WS_OUTPTS;
  float* tgt_pts = ws + WS_TGTPTS;
  float* partial = ws + WS_PARTIAL;
  float* out     = (float*)d_out;

  softmax_kernel<<<(B_ * QPAD_ + 255) / 256, 256, 0, stream>>>(pred_logits, probs);

  long long totP = (long long)B_ * Q_ * P_;
  sample_kernel<<<(unsigned)((totP + 255) / 256), 256, 0, stream>>>(
      pred_masks, coords, out_pts, Q_, HP_, WP_, totP);

  long long totT = (long long)B_ * T_ * P_;
  sample_kernel<<<(unsigned)((totT + 255) / 256), 256, 0, stream>>>(
      tgt_masks, coords, tgt_pts, T_, HG_, WG_, totT);

  class_cost_kernel<<<B_ * 16 * 2, 32, 0, stream>>>(probs, labels, classc);

  pair_l1_kernel<<<NS_ * B_ * 16, 256, 0, stream>>>(out_pts, tgt_pts, partial);

  finalize_kernel<<<(B_ * Q_ * T_ + 255) / 256, 256, 0, stream>>>(partial, classc, out);
}